// ContrastHead_33517924778311
// MI455X (gfx1250) — compile-verified
//
#include <hip/hip_runtime.h>
#include <math.h>

typedef __attribute__((ext_vector_type(2))) float v2f;
typedef __attribute__((ext_vector_type(8))) float v8f;

#define C_DIM   64
#define K_NBR   35
#define NTILES  3            // ceil(35/16)
#define TEMP_INV 10.0f       // 1 / 0.1
#define EPS_F    1e-7f

// sum across each 16-lane half (wave32); halves hold mirrored data
__device__ __forceinline__ float hsum16(float v) {
    v += __shfl_xor(v, 1, 32);
    v += __shfl_xor(v, 2, 32);
    v += __shfl_xor(v, 4, 32);
    v += __shfl_xor(v, 8, 32);
    return v;
}
__device__ __forceinline__ float hmax16(float v) {
    v = fmaxf(v, __shfl_xor(v, 1, 32));
    v = fmaxf(v, __shfl_xor(v, 2, 32));
    v = fmaxf(v, __shfl_xor(v, 4, 32));
    v = fmaxf(v, __shfl_xor(v, 8, 32));
    return v;
}
__device__ __forceinline__ float hsum32(float v) {
    v = hsum16(v);
    v += __shfl_xor(v, 16, 32);
    return v;
}

__global__ void __launch_bounds__(128)
contrast_main(const float* __restrict__ feat,
              const int*   __restrict__ labels,
              const int*   __restrict__ nidx,
              int M, float* __restrict__ partials)
{
    const int lane   = threadIdx.x & 31;
    const int wave   = (blockIdx.x * blockDim.x + threadIdx.x) >> 5;
    const int nwaves = (gridDim.x * blockDim.x) >> 5;
    const int col    = lane & 15;   // column / row index inside 16-tile
    const int half   = lane >> 4;   // 0: K={0,1}, 1: K={2,3} of each 4-chunk

    float accLoss = 0.0f, accCnt = 0.0f;

    for (int p = wave; p < M; p += nwaves) {
        // prefetch next point's neighbor-index row into cache
        if (p + nwaves < M)
            __builtin_prefetch(nidx + (size_t)(p + nwaves) * K_NBR, 0, 1);

        // ---- own features: lane l holds f[2l], f[2l+1] (coalesced b64) ----
        v2f fi = *(const v2f*)(feat + (size_t)p * C_DIM + 2 * lane);
        float normI = hsum32(fi.x * fi.x + fi.y * fi.y);
        int myLab = labels[p];

        // ---- build A operand: f_i replicated into all 16 rows ----
        // A 16x4 f32 layout: lanes 0-15 hold K0(V0),K1(V1); lanes 16-31 K2,K3
        v2f a[16];
#pragma unroll
        for (int t = 0; t < 16; ++t) {
            int src = 2 * t + half;           // lane owning f[4t + 2*half]
            a[t].x = __shfl(fi.x, src, 32);
            a[t].y = __shfl(fi.y, src, 32);
        }

        float dd[NTILES];      // -dist per (tile, col)
        float nrm[NTILES];     // neighbor norms
        int   labk[NTILES];
        bool  validk[NTILES];

#pragma unroll
        for (int t = 0; t < NTILES; ++t) {
            int  row   = t * 16 + col;
            bool valid = (row < K_NBR);
            int  rowc  = valid ? row : (K_NBR - 1);
            int  n     = nidx[(size_t)p * K_NBR + rowc];
            labk[t]    = labels[n];
            validk[t]  = valid;

            // B operand: column col = neighbor row n; lane loads its K-pair
            const float* nf = feat + (size_t)n * C_DIM + half * 2;
            v8f acc = {0.f, 0.f, 0.f, 0.f, 0.f, 0.f, 0.f, 0.f};
            float bn = 0.0f;
#pragma unroll
            for (int c4 = 0; c4 < 16; ++c4) {
                v2f b = *(const v2f*)(nf + 4 * c4);
                bn += b.x * b.x + b.y * b.y;
                acc = __builtin_amdgcn_wmma_f32_16x16x4_f32(
                        false, a[c4], false, b, (short)0, acc, false, false);
            }
            // all A rows identical -> every D row equals dot(f_i, n_col):
            float dot = acc[0];
            float nN  = bn + __shfl_xor(bn, 16, 32);   // combine K-halves
            nrm[t] = nN;
            float d2   = normI + nN - 2.0f * dot;
            float dist = sqrtf(fmaxf(d2, 0.0f) + EPS_F);
            dd[t] = -dist;
        }

        // ---- softmax over the 35 neighbors (lane-aligned, 3 regs/lane) ----
        float m = -1e30f;
#pragma unroll
        for (int t = 0; t < NTILES; ++t)
            m = validk[t] ? fmaxf(m, dd[t]) : m;
        m = hmax16(m);

        float pos = 0.f, neg = 0.f, cnt = 0.f;
#pragma unroll
        for (int t = 0; t < NTILES; ++t) {
            float e = validk[t] ? __expf((dd[t] - m) * TEMP_INV) : 0.0f;
            bool  f = validk[t] && (labk[t] == myLab);
            neg += e;
            pos += f ? e : 0.0f;
            cnt += f ? 1.0f : 0.0f;
        }
        pos = hsum16(pos);
        neg = hsum16(neg);
        cnt = hsum16(cnt);

        float loss = -logf(pos / neg + EPS_F);
        float pm   = (cnt > 0.5f && cnt < (float)K_NBR - 0.5f) ? 1.0f : 0.0f;
        accLoss += loss * pm;
        accCnt  += pm;
    }

    if (lane == 0) {
        partials[2 * wave + 0] = accLoss;
        partials[2 * wave + 1] = accCnt;
    }
}

__global__ void contrast_final(const float* __restrict__ partials,
                               int nwaves, float* __restrict__ out)
{
    float s = 0.f, c = 0.f;
    for (int i = threadIdx.x; i < nwaves; i += 32) {
        s += partials[2 * i + 0];
        c += partials[2 * i + 1];
    }
    s = hsum32(s);
    c = hsum32(c);
    if (threadIdx.x == 0)
        out[0] = s / fmaxf(c, 1.0f);   // WEIGHT == 1.0
}

extern "C" void kernel_launch(void* const* d_in, const int* in_sizes, int n_in,
                              void* d_out, int out_size, void* d_ws, size_t ws_size,
                              hipStream_t stream)
{
    const float* feat   = (const float*)d_in[0];
    const int*   labels = (const int*)d_in[1];
    const int*   nidx   = (const int*)d_in[2];
    const int M = in_sizes[1];

    const int threads = 128;
    int blocks = 1024;
    // cap by workspace capacity (2 floats per wave), deterministic per ws_size
    while (blocks > 1 &&
           (size_t)(blocks * threads / 32) * 2 * sizeof(float) > ws_size)
        blocks >>= 1;
    const int nwaves = blocks * threads / 32;

    contrast_main<<<blocks, threads, 0, stream>>>(feat, labels, nidx, M,
                                                  (float*)d_ws);
    contrast_final<<<1, 32, 0, stream>>>((const float*)d_ws, nwaves,
                                         (float*)d_out);
}